// Self_Attention_23356032156209
// MI455X (gfx1250) — compile-verified
//
#include <hip/hip_runtime.h>
#include <hip/hip_bf16.h>
#include <math.h>

// ---------------------------------------------------------------------------
// Problem constants (match reference)
// ---------------------------------------------------------------------------
#define GN      4096      // nodes
#define GE      131072    // edges
#define GF      512       // in_feats == d_k == d_v

typedef float v2f __attribute__((ext_vector_type(2)));
typedef float v8f __attribute__((ext_vector_type(8)));

// ---------------------------------------------------------------------------
// Small utility kernels
// ---------------------------------------------------------------------------
__global__ void zero_kernel(float* p, int n) {
    int i = blockIdx.x * blockDim.x + threadIdx.x;
    if (i < n) p[i] = 0.0f;
}

__global__ void degree_kernel(const int* __restrict__ edge_index,
                              float* __restrict__ deg_src,
                              float* __restrict__ deg_dst) {
    int e = blockIdx.x * blockDim.x + threadIdx.x;
    if (e < GE) {
        int s = edge_index[e];        // row 0: src
        int d = edge_index[GE + e];   // row 1: dst
        atomicAdd(&deg_src[s], 1.0f);
        atomicAdd(&deg_dst[d], 1.0f);
    }
}

__global__ void norm_finalize_kernel(float* p, int n) {
    int i = blockIdx.x * blockDim.x + threadIdx.x;
    if (i < n) {
        float v = p[i];
        v = v < 1.0f ? 1.0f : v;
        p[i] = rsqrtf(v);
    }
}

// out[i][f] = in[i][f] * norm[i]   (in == out allowed)
__global__ void row_scale_kernel(const float* __restrict__ in,
                                 const float* __restrict__ norm,
                                 float* __restrict__ out) {
    int i = blockIdx.x * blockDim.x + threadIdx.x;   // over N*GF
    if (i < GN * GF) {
        int row = i >> 9;  // /512
        out[i] = in[i] * norm[row];
    }
}

// Scatter-add along edges: agg[dst] += h[src].  One thread = (edge, 4 feats).
__global__ void scatter_kernel(const int* __restrict__ edge_index,
                               const float* __restrict__ h,
                               float* __restrict__ agg) {
    long long t = (long long)blockIdx.x * blockDim.x + threadIdx.x;
    const long long total = (long long)GE * (GF / 4);
    if (t < total) {
        int e  = (int)(t >> 7);        // /128
        int f4 = (int)(t & 127);
        int s = edge_index[e];
        int d = edge_index[GE + e];
        const float4 v = *reinterpret_cast<const float4*>(&h[(long long)s * GF + f4 * 4]);
        float* ap = &agg[(long long)d * GF + f4 * 4];
        atomicAdd(ap + 0, v.x);
        atomicAdd(ap + 1, v.y);
        atomicAdd(ap + 2, v.z);
        atomicAdd(ap + 3, v.w);
    }
}

// ---------------------------------------------------------------------------
// FP32 WMMA GEMM:  C[M,N] = alpha * A @ op(B) (+ bias[n]) (+ addm[m,n])
//   TRANSB=0: B is [K,N] row-major.   TRANSB=1: B is [N,K] row-major (B^T).
//
// 256 threads (8 waves), 128x128 block tile, BK=16, double-buffered LDS.
// Waves in 4(M) x 2(N); each wave owns 32x64 = 2x4 v8f accumulators.
// A tile: row pitch 20 (16B-aligned b128 staging, conflict-free b64 frag reads)
// B tile: K-pair interleaved, pair-group pitch 264 -> every A/B fragment is
//         one contiguous aligned ds_load_b64 (no packing movs).
// Staging loads are float4 (global_load_b128) prefetched one tile ahead.
// __launch_bounds__(256, 2): cap occupancy at 2 waves/EU so the ~150 live
// VGPRs (64 accum + staging + addresses) fit without scratch spills.
// ---------------------------------------------------------------------------
template <bool TRANSB, bool HAS_BIAS, bool HAS_ADD>
__global__ __launch_bounds__(256, 2)
void gemm_wmma_f32(const float* __restrict__ A,
                   const float* __restrict__ B,
                   const float* __restrict__ bias,
                   const float* __restrict__ addm,
                   float* __restrict__ C,
                   int M, int N, int K, float alpha) {
    constexpr int AP = 20;    // A row pitch (floats)
    constexpr int BP = 264;   // B pair-group pitch (floats)
    __shared__ float As[2][128 * AP];
    __shared__ float Bs[2][8 * BP];

    const int t      = threadIdx.x;
    const int lane   = t & 31;
    const int waveId = t >> 5;           // 0..7
    const int wm     = waveId >> 1;      // 0..3  -> M offset 32*wm
    const int wn     = waveId & 1;       // 0..1  -> N offset 64*wn
    const int lo     = lane & 15;
    const int hi     = lane >> 4;

    const int m0 = blockIdx.y * 128;
    const int n0 = blockIdx.x * 128;

    v8f c[2][4];
    #pragma unroll
    for (int i = 0; i < 2; ++i)
        #pragma unroll
        for (int j = 0; j < 4; ++j)
            c[i][j] = (v8f){};

    float4 ra[2], rb[2];

    // per-thread staging coordinates (tile has 512 float4s; 2 per thread)
    // A: q -> row q>>2, col4 q&3
    // B !TRANSB: q -> k q>>5, n4 q&31     (reads coalesced along n)
    // B  TRANSB: q -> n q>>2, k4 q&3      (reads coalesced along k)
    const int q0 = t, q1 = t + 256;

    auto load_tile = [&](int k0) {
        ra[0] = *reinterpret_cast<const float4*>(
                    &A[(long long)(m0 + (q0 >> 2)) * K + k0 + (q0 & 3) * 4]);
        ra[1] = *reinterpret_cast<const float4*>(
                    &A[(long long)(m0 + (q1 >> 2)) * K + k0 + (q1 & 3) * 4]);
        if (TRANSB) {
            rb[0] = *reinterpret_cast<const float4*>(
                        &B[(long long)(n0 + (q0 >> 2)) * K + k0 + (q0 & 3) * 4]);
            rb[1] = *reinterpret_cast<const float4*>(
                        &B[(long long)(n0 + (q1 >> 2)) * K + k0 + (q1 & 3) * 4]);
        } else {
            rb[0] = *reinterpret_cast<const float4*>(
                        &B[(long long)(k0 + (q0 >> 5)) * N + n0 + (q0 & 31) * 4]);
            rb[1] = *reinterpret_cast<const float4*>(
                        &B[(long long)(k0 + (q1 >> 5)) * N + n0 + (q1 & 31) * 4]);
        }
    };

    auto store_tile = [&](int buf) {
        *reinterpret_cast<float4*>(&As[buf][(q0 >> 2) * AP + (q0 & 3) * 4]) = ra[0];
        *reinterpret_cast<float4*>(&As[buf][(q1 >> 2) * AP + (q1 & 3) * 4]) = ra[1];
        if (TRANSB) {
            #pragma unroll
            for (int i = 0; i < 2; ++i) {
                const int q  = i ? q1 : q0;
                const int n  = q >> 2, k4 = q & 3;
                const float4 v = rb[i];
                *reinterpret_cast<v2f*>(&Bs[buf][(2 * k4 + 0) * BP + 2 * n]) = (v2f){v.x, v.y};
                *reinterpret_cast<v2f*>(&Bs[buf][(2 * k4 + 1) * BP + 2 * n]) = (v2f){v.z, v.w};
            }
        } else {
            #pragma unroll
            for (int i = 0; i < 2; ++i) {
                const int q  = i ? q1 : q0;
                const int k  = q >> 5, n4 = q & 31;
                const int base = (k >> 1) * BP + (k & 1);
                const float4 v = rb[i];
                Bs[buf][base + (n4 * 4 + 0) * 2] = v.x;
                Bs[buf][base + (n4 * 4 + 1) * 2] = v.y;
                Bs[buf][base + (n4 * 4 + 2) * 2] = v.z;
                Bs[buf][base + (n4 * 4 + 3) * 2] = v.w;
            }
        }
    };

    const int nt = K >> 4;
    load_tile(0);

    for (int kt = 0; kt < nt; ++kt) {
        const int cur = kt & 1;
        store_tile(cur);
        __syncthreads();
        if (kt + 1 < nt) load_tile((kt + 1) << 4);   // prefetch overlaps WMMA

        const float* as = As[cur];
        const float* bs = Bs[cur];
        #pragma unroll
        for (int kk = 0; kk < 4; ++kk) {
            const int kb = kk * 4 + 2 * hi;          // this lane-group's K pair
            v2f a[2], b[4];
            #pragma unroll
            for (int i = 0; i < 2; ++i)
                a[i] = *reinterpret_cast<const v2f*>(
                           &as[(wm * 32 + i * 16 + lo) * AP + kb]);
            #pragma unroll
            for (int j = 0; j < 4; ++j)
                b[j] = *reinterpret_cast<const v2f*>(
                           &bs[(kk * 2 + hi) * BP + (wn * 64 + j * 16 + lo) * 2]);
            #pragma unroll
            for (int i = 0; i < 2; ++i)
                #pragma unroll
                for (int j = 0; j < 4; ++j)
                    c[i][j] = __builtin_amdgcn_wmma_f32_16x16x4_f32(
                                  false, a[i], false, b[j], (short)0, c[i][j],
                                  false, false);
        }
        // single barrier per tile: next iteration stores into the other buffer
    }

    // --- epilogue: VGPR r of a frag holds rows (r, r+8) across lane halves ---
    #pragma unroll
    for (int i = 0; i < 2; ++i) {
        const int rbase = m0 + wm * 32 + i * 16 + 8 * hi;
        #pragma unroll
        for (int j = 0; j < 4; ++j) {
            const int col = n0 + wn * 64 + j * 16 + lo;
            #pragma unroll
            for (int r = 0; r < 8; ++r) {
                const int row = rbase + r;
                float v = c[i][j][r] * alpha;
                if (HAS_BIAS) v += bias[col];
                if (HAS_ADD)  v += addm[(long long)row * N + col];
                C[(long long)row * N + col] = v;
            }
        }
    }
}

// ---------------------------------------------------------------------------
// Row softmax over [GN, GN], in place. One 256-thread block per row.
// ---------------------------------------------------------------------------
__global__ __launch_bounds__(256)
void softmax_rows_kernel(float* __restrict__ s) {
    __shared__ float red[256];
    const int row = blockIdx.x;
    float* p = s + (long long)row * GN;
    const int t = threadIdx.x;

    float m = -1e30f;
    for (int i = t; i < GN; i += 256) m = fmaxf(m, p[i]);
    red[t] = m; __syncthreads();
    for (int o = 128; o > 0; o >>= 1) {
        if (t < o) red[t] = fmaxf(red[t], red[t + o]);
        __syncthreads();
    }
    m = red[0]; __syncthreads();

    float acc = 0.0f;
    for (int i = t; i < GN; i += 256) {
        float e = __expf(p[i] - m);
        p[i] = e;
        acc += e;
    }
    red[t] = acc; __syncthreads();
    for (int o = 128; o > 0; o >>= 1) {
        if (t < o) red[t] += red[t + o];
        __syncthreads();
    }
    const float inv = 1.0f / red[0];
    __syncthreads();

    for (int i = t; i < GN; i += 256) p[i] *= inv;
}

// ---------------------------------------------------------------------------
// Host-side orchestration
// ---------------------------------------------------------------------------
extern "C" void kernel_launch(void* const* d_in, const int* in_sizes, int n_in,
                              void* d_out, int out_size, void* d_ws, size_t ws_size,
                              hipStream_t stream) {
    (void)in_sizes; (void)n_in; (void)out_size; (void)ws_size;

    const float* x    = (const float*)d_in[0];   // [N, F]
    const int*   ei   = (const int*)  d_in[1];   // [2, E]
    const float* hatt = (const float*)d_in[2];   // [N, N]
    const float* W_Q  = (const float*)d_in[3];
    const float* b_Q  = (const float*)d_in[4];
    const float* W_K  = (const float*)d_in[5];
    const float* b_K  = (const float*)d_in[6];
    const float* W_V  = (const float*)d_in[7];
    const float* b_V  = (const float*)d_in[8];
    const float* W_O  = (const float*)d_in[9];
    const float* b_O  = (const float*)d_in[10];

    float* out_attnout = (float*)d_out;                         // [N, F]
    float* attn        = (float*)d_out + (long long)GN * GF;    // [N, N]

    // workspace layout (floats)
    float* ws        = (float*)d_ws;
    float* norm_src  = ws;                         // N
    float* norm_dst  = ws + GN;                    // N
    float* hbuf      = ws + 2 * GN;                // N*F
    float* agg       = hbuf + (long long)GN * GF;  // N*F
    float* Q         = agg  + (long long)GN * GF;  // N*F
    float* Km        = Q    + (long long)GN * GF;  // N*F
    float* V         = Km   + (long long)GN * GF;  // N*F
    float* ao0       = V    + (long long)GN * GF;  // N*F (attn @ V)

    const int NF = GN * GF;
    const float scale = 1.0f / sqrtf((float)GF);

    // ---- degrees -> norms (shared by first and last graph_conv) ----
    zero_kernel<<<(2 * GN + 255) / 256, 256, 0, stream>>>(norm_src, 2 * GN);
    degree_kernel<<<(GE + 255) / 256, 256, 0, stream>>>(ei, norm_src, norm_dst);
    norm_finalize_kernel<<<(2 * GN + 255) / 256, 256, 0, stream>>>(norm_src, 2 * GN);

    // ---- shared aggregation: agg = Ddst^-1/2 A Dsrc^-1/2 X ----
    row_scale_kernel<<<(NF + 255) / 256, 256, 0, stream>>>(x, norm_src, hbuf);
    zero_kernel<<<(NF + 255) / 256, 256, 0, stream>>>(agg, NF);
    {
        long long total = (long long)GE * (GF / 4);
        scatter_kernel<<<(unsigned)((total + 255) / 256), 256, 0, stream>>>(ei, hbuf, agg);
    }
    row_scale_kernel<<<(NF + 255) / 256, 256, 0, stream>>>(agg, norm_dst, agg);

    // ---- Q/K/V = agg @ W + b   (M=4096, N=512, K=512) ----
    dim3 gQ(GF / 128, GN / 128);
    gemm_wmma_f32<false, true, false><<<gQ, 256, 0, stream>>>(agg, W_Q, b_Q, nullptr, Q,  GN, GF, GF, 1.0f);
    gemm_wmma_f32<false, true, false><<<gQ, 256, 0, stream>>>(agg, W_K, b_K, nullptr, Km, GN, GF, GF, 1.0f);
    gemm_wmma_f32<false, true, false><<<gQ, 256, 0, stream>>>(agg, W_V, b_V, nullptr, V,  GN, GF, GF, 1.0f);

    // ---- scores = Q @ K^T * scale + h_attn   -> attn slice of d_out ----
    dim3 gS(GN / 128, GN / 128);
    gemm_wmma_f32<true, false, true><<<gS, 256, 0, stream>>>(Q, Km, nullptr, hatt, attn, GN, GN, GF, scale);

    // ---- softmax rows, in place ----
    softmax_rows_kernel<<<GN, 256, 0, stream>>>(attn);

    // ---- attn_out0 = attn @ V  (M=4096, N=512, K=4096) ----
    gemm_wmma_f32<false, false, false><<<gQ, 256, 0, stream>>>(attn, V, nullptr, nullptr, ao0, GN, GF, GN, 1.0f);

    // ---- final graph_conv on attn_out0 ----
    row_scale_kernel<<<(NF + 255) / 256, 256, 0, stream>>>(ao0, norm_src, hbuf);
    zero_kernel<<<(NF + 255) / 256, 256, 0, stream>>>(agg, NF);
    {
        long long total = (long long)GE * (GF / 4);
        scatter_kernel<<<(unsigned)((total + 255) / 256), 256, 0, stream>>>(ei, hbuf, agg);
    }
    row_scale_kernel<<<(NF + 255) / 256, 256, 0, stream>>>(agg, norm_dst, agg);
    gemm_wmma_f32<false, true, false><<<gQ, 256, 0, stream>>>(agg, W_O, b_O, nullptr, out_attnout, GN, GF, GF, 1.0f);
}